// GraphGATAggregator_31413390803232
// MI455X (gfx1250) — compile-verified
//
#include <hip/hip_runtime.h>
#include <hip/hip_bf16.h>
#include <math.h>

typedef __attribute__((ext_vector_type(2))) float v2f;
typedef __attribute__((ext_vector_type(8))) float v8f;

#define D 64

// ---------------------------------------------------------------------------
// Kernel 1: Wh = X @ W^T  using V_WMMA_F32_16X16X4_F32 (fp32 WMMA, exact math)
// One wave computes a 16-row x 64-col block: 4 accumulator tiles, 16 K-steps.
// A (16x4 f32) layout: lanes 0-15 rows, VGPR0/1 = K0/K1; lanes 16-31 = K2/K3.
// B (4x16)  layout mirrors A with N across lanes; B[k][n] = W[n][k].
// C/D (16x16 f32): lane<16 -> rows v (v=0..7), lane>=16 -> rows v+8; col = lane%16.
// ---------------------------------------------------------------------------
__global__ __launch_bounds__(256) void gemm_wh_kernel(const float* __restrict__ X,
                                                      const float* __restrict__ W,
                                                      float* __restrict__ Wh, int N) {
  const int lane = threadIdx.x & 31;
  const int wave = (blockIdx.x << 3) + (threadIdx.x >> 5);
  const int row0 = wave << 4;                 // 16 rows per wave
  if (row0 >= N) return;
  const int half = lane >> 4;                 // 0: lanes 0-15, 1: lanes 16-31
  const int l16  = lane & 15;

  if (row0 + 16 <= N) {
    v8f acc0 = {}, acc1 = {}, acc2 = {}, acc3 = {};
    #pragma unroll
    for (int k = 0; k < D; k += 4) {
      const float* arow = X + (size_t)(row0 + l16) * D + k + half * 2;
      v2f a; a.x = arow[0]; a.y = arow[1];
      const float* w0 = W + (size_t)(0 * 16 + l16) * D + k + half * 2;
      const float* w1 = W + (size_t)(1 * 16 + l16) * D + k + half * 2;
      const float* w2 = W + (size_t)(2 * 16 + l16) * D + k + half * 2;
      const float* w3 = W + (size_t)(3 * 16 + l16) * D + k + half * 2;
      v2f b0; b0.x = w0[0]; b0.y = w0[1];
      v2f b1; b1.x = w1[0]; b1.y = w1[1];
      v2f b2; b2.x = w2[0]; b2.y = w2[1];
      v2f b3; b3.x = w3[0]; b3.y = w3[1];
      acc0 = __builtin_amdgcn_wmma_f32_16x16x4_f32(false, a, false, b0, (short)0, acc0, false, false);
      acc1 = __builtin_amdgcn_wmma_f32_16x16x4_f32(false, a, false, b1, (short)0, acc1, false, false);
      acc2 = __builtin_amdgcn_wmma_f32_16x16x4_f32(false, a, false, b2, (short)0, acc2, false, false);
      acc3 = __builtin_amdgcn_wmma_f32_16x16x4_f32(false, a, false, b3, (short)0, acc3, false, false);
    }
    #pragma unroll
    for (int v = 0; v < 8; ++v) {
      const size_t r = (size_t)(row0 + v + half * 8) * D;
      Wh[r + 0 * 16 + l16] = acc0[v];
      Wh[r + 1 * 16 + l16] = acc1[v];
      Wh[r + 2 * 16 + l16] = acc2[v];
      Wh[r + 3 * 16 + l16] = acc3[v];
    }
  } else {
    // scalar tail (only if N % 16 != 0); wave-uniform branch keeps WMMA path EXEC=all-1s
    for (int r = row0 + lane; r < N; r += 32) {
      for (int j = 0; j < D; ++j) {
        float s = 0.f;
        for (int k = 0; k < D; ++k) s += X[(size_t)r * D + k] * W[(size_t)j * D + k];
        Wh[(size_t)r * D + j] = s;
      }
    }
  }
}

// ---------------------------------------------------------------------------
// Kernel 2: s_src[i] = Wh[i].a[:64], s_dst[i] = Wh[i].a[64:128]
// One wave per node; float2 per lane; butterfly reduce.
// ---------------------------------------------------------------------------
__global__ __launch_bounds__(256) void scores_kernel(const float* __restrict__ Wh,
                                                     const float* __restrict__ a,
                                                     float* __restrict__ s_src,
                                                     float* __restrict__ s_dst, int N) {
  const int node = (blockIdx.x << 3) + (threadIdx.x >> 5);
  if (node >= N) return;
  const int lane = threadIdx.x & 31;
  const float2 x  = ((const float2*)(Wh + (size_t)node * D))[lane];
  const float2 as = ((const float2*)a)[lane];
  const float2 ad = ((const float2*)(a + D))[lane];
  float ps = x.x * as.x + x.y * as.y;
  float pd = x.x * ad.x + x.y * ad.y;
  #pragma unroll
  for (int off = 16; off > 0; off >>= 1) {
    ps += __shfl_xor(ps, off, 32);
    pd += __shfl_xor(pd, off, 32);
  }
  if (lane == 0) { s_src[node] = ps; s_dst[node] = pd; }
}

// ---------------------------------------------------------------------------
// Kernel 3: CSR row offsets from sorted edge_src (lower_bound per node).
// ---------------------------------------------------------------------------
__global__ __launch_bounds__(256) void row_offsets_kernel(const int* __restrict__ edge_src,
                                                          int E, int N, int* __restrict__ rs) {
  const int u = blockIdx.x * 256 + threadIdx.x;
  if (u > N) return;
  int lo = 0, hi = E;
  while (lo < hi) {
    int mid = (lo + hi) >> 1;
    if (edge_src[mid] < u) lo = mid + 1; else hi = mid;
  }
  rs[u] = lo;
}

// ---------------------------------------------------------------------------
// Kernel 4: per-node segment softmax + weighted aggregation. One wave / node.
// Pass 1: lanes over edges (max, denom). Pass 2: edges sequential, lanes over
// the 64 features -> each edge is one fully coalesced 256B Wh row gather.
// ---------------------------------------------------------------------------
__global__ __launch_bounds__(256) void aggregate_kernel(const float* __restrict__ Wh,
                                                        const float* __restrict__ s_src,
                                                        const float* __restrict__ s_dst,
                                                        const int* __restrict__ edge_dst,
                                                        const int* __restrict__ rs,
                                                        float* __restrict__ out, int N) {
  const int u = (blockIdx.x << 3) + (threadIdx.x >> 5);
  if (u >= N) return;
  const int lane = threadIdx.x & 31;
  const int start = rs[u];
  const int end   = rs[u + 1];

  if (end == start) {                 // no outgoing edges: relu(Wh[u])
    float2 w = ((const float2*)(Wh + (size_t)u * D))[lane];
    float2 r; r.x = fmaxf(w.x, 0.f); r.y = fmaxf(w.y, 0.f);
    ((float2*)(out + (size_t)u * D))[lane] = r;
    return;
  }

  const float ssrc = s_src[u];

  // pass 1a: segment max of leaky_relu scores
  float m = -__builtin_inff();
  for (int i = start + lane; i < end; i += 32) {
    float e = ssrc + s_dst[edge_dst[i]];
    e = (e > 0.f) ? e : 0.2f * e;
    m = fmaxf(m, e);
  }
  #pragma unroll
  for (int off = 16; off > 0; off >>= 1) m = fmaxf(m, __shfl_xor(m, off, 32));

  // pass 1b: denom = sum exp(e - m)
  float denom = 0.f;
  for (int i = start + lane; i < end; i += 32) {
    float e = ssrc + s_dst[edge_dst[i]];
    e = (e > 0.f) ? e : 0.2f * e;
    denom += __expf(e - m);
  }
  #pragma unroll
  for (int off = 16; off > 0; off >>= 1) denom += __shfl_xor(denom, off, 32);
  const float inv = 1.f / denom;

  // pass 2: acc += alpha * Wh[dst]; lanes cover the 64 features via float2
  float acc_x = 0.f, acc_y = 0.f;
  for (int i = start; i < end; ++i) {
    const int v = edge_dst[i];                    // wave-uniform -> scalar load
    float e = ssrc + s_dst[v];
    e = (e > 0.f) ? e : 0.2f * e;
    const float alpha = __expf(e - m) * inv;
    const float2 w = ((const float2*)(Wh + (size_t)v * D))[lane];
    acc_x += alpha * w.x;
    acc_y += alpha * w.y;
  }
  float2 r; r.x = fmaxf(acc_x, 0.f); r.y = fmaxf(acc_y, 0.f);
  ((float2*)(out + (size_t)u * D))[lane] = r;
}

// ---------------------------------------------------------------------------
extern "C" void kernel_launch(void* const* d_in, const int* in_sizes, int n_in,
                              void* d_out, int out_size, void* d_ws, size_t ws_size,
                              hipStream_t stream) {
  const float* X    = (const float*)d_in[0];   // user_states [N, 64]
  const float* W    = (const float*)d_in[1];   // W [64, 64]
  const float* a    = (const float*)d_in[2];   // a [128]
  const int*   esrc = (const int*)d_in[3];     // edge_src [E] (sorted)
  const int*   edst = (const int*)d_in[4];     // edge_dst [E]
  const int N = in_sizes[0] / D;
  const int E = in_sizes[3];

  char* ws = (char*)d_ws;
  size_t off = 0;
  auto carve = [&](size_t bytes) -> void* {
    void* p = ws + off;
    off = (off + bytes + 255) & ~(size_t)255;
    return p;
  };
  float* Wh    = (float*)carve((size_t)N * D * sizeof(float));
  float* s_src = (float*)carve((size_t)N * sizeof(float));
  float* s_dst = (float*)carve((size_t)N * sizeof(float));
  int*   rs    = (int*)carve((size_t)(N + 1) * sizeof(int));
  float* out   = (float*)d_out;

  const int mtiles = (N + 15) / 16;                 // 16 rows per wave
  gemm_wh_kernel<<<(mtiles + 7) / 8, 256, 0, stream>>>(X, W, Wh, N);
  scores_kernel<<<(N + 7) / 8, 256, 0, stream>>>(Wh, a, s_src, s_dst, N);
  row_offsets_kernel<<<(N + 1 + 255) / 256, 256, 0, stream>>>(esrc, E, N, rs);
  aggregate_kernel<<<(N + 7) / 8, 256, 0, stream>>>(Wh, s_src, s_dst, edst, rs, out, N);
}